// MiniBatch_17824114279178
// MI455X (gfx1250) — compile-verified
//
#include <hip/hip_runtime.h>
#include <math.h>

typedef __attribute__((ext_vector_type(2))) float v2f;
typedef __attribute__((ext_vector_type(8))) float v8f;

#define B_        512
#define INF_      512
#define OUTF_     64
#define KD_       16
#define NCOL_     1024     // OUTF_ * KD_
#define OUTSTR_   577      // INF_ + OUTF_ + 1
#define SPAD_     17       // LDS row stride (pad to avoid bank conflicts)
#define NCHUNK_   8        // i-dimension split for the pairwise pass
#define CHUNK_    (B_ / NCHUNK_)   // 64 rows per chunk

// ---------------------------------------------------------------------------
// Kernel 1: M[f][i][k] = sum_c x[i][c] * T[c][f*16 + k]   via f32 WMMA
// One wave computes one 16(i) x 16(k) tile for one out-feature f.
// 2048 tiles total -> 256 blocks x 8 waves.
// ---------------------------------------------------------------------------
__global__ __launch_bounds__(256) void gemm_wmma_kernel(
    const float* __restrict__ x, const float* __restrict__ T,
    float* __restrict__ Mw) {
  const int lane = threadIdx.x & 31;
  const int wave = threadIdx.x >> 5;
  const int tile = blockIdx.x * 8 + wave;   // 0..2047
  const int f    = tile & 63;               // N-tile == out-feature (KD_==16)
  const int i0   = (tile >> 6) << 4;        // batch-row tile base
  const int half = lane >> 4;               // 0: K={0,1}, 1: K={2,3} of each step
  const int l    = lane & 15;

  // A (16x4, f32): lane l holds row i0+l; VGPR pair = two consecutive K.
  const float* aptr = x + (size_t)(i0 + l) * INF_ + 2 * half;
  // B (4x16, f32): lane l holds col f*16+l; rows k0+2*half, k0+2*half+1.
  const float* bptr = T + f * KD_ + l;

  v8f acc = {};
#pragma unroll 4
  for (int k0 = 0; k0 < INF_; k0 += 4) {
    v2f a = *(const v2f*)(aptr + k0);                 // global_load_b64
    v2f b;
    b.x = bptr[(size_t)(k0 + 2 * half) * NCOL_];
    b.y = bptr[(size_t)(k0 + 2 * half + 1) * NCOL_];
    // (neg_a, A, neg_b, B, c_mod, C, reuse_a, reuse_b)
    acc = __builtin_amdgcn_wmma_f32_16x16x4_f32(false, a, false, b,
                                                (short)0, acc, false, false);
  }

  // C layout: VGPR r, lanes 0-15 -> row r, lanes 16-31 -> row r+8; col = l.
  float* dst = Mw + (size_t)f * (B_ * KD_) + (size_t)i0 * KD_ + l;
#pragma unroll
  for (int r = 0; r < 8; ++r) {
    const int row = half ? (r + 8) : r;
    dst[row * KD_] = acc[r];
  }
}

// ---------------------------------------------------------------------------
// Kernel 2: partial sim sums.
// Grid = 64 f x 8 i-chunks = 512 blocks (fills the WGP array), 512 threads.
// Block (f, c): part[c][f][j] = sum_{i in chunk c} exp(-L1(M[i,f,:], M[j,f,:]))
// Chunk slice (64x16 f32 = 4KB) staged in padded LDS; mj read from L2.
// ---------------------------------------------------------------------------
__global__ __launch_bounds__(512) void sim_part_kernel(
    const float* __restrict__ Mw, float* __restrict__ part) {
  __shared__ float sM[CHUNK_ * SPAD_];
  const int f     = blockIdx.x & 63;
  const int chunk = blockIdx.x >> 6;           // 0..7
  const float* Mf = Mw + (size_t)f * (B_ * KD_);
  const int i0    = chunk * CHUNK_;

  for (int idx = threadIdx.x; idx < CHUNK_ * KD_; idx += 512) {
    const int row = idx >> 4, k = idx & 15;
    sM[row * SPAD_ + k] = Mf[(size_t)(i0 + row) * KD_ + k];
  }
  __syncthreads();

  const int j = threadIdx.x;
  float mj[KD_];
#pragma unroll
  for (int k = 0; k < KD_; ++k) mj[k] = Mf[(size_t)j * KD_ + k];

  float acc = 0.f;
  for (int i = 0; i < CHUNK_; ++i) {
    float d = 0.f;
#pragma unroll
    for (int k = 0; k < KD_; ++k) d += fabsf(sM[i * SPAD_ + k] - mj[k]);
    acc += __expf(-d);
  }
  part[(size_t)blockIdx.x * B_ + j] = acc;
}

// ---------------------------------------------------------------------------
// Kernel 3: deterministic reduction of the 8 chunk partials (+ the -1
// self-similarity correction) into the output sim columns.
// 32768 outputs -> 128 blocks x 256 threads.
// ---------------------------------------------------------------------------
__global__ __launch_bounds__(256) void sim_reduce_kernel(
    const float* __restrict__ part, float* __restrict__ out) {
  const int idx = blockIdx.x * 256 + threadIdx.x;  // idx = f*512 + j
  const int f = idx >> 9;
  const int j = idx & 511;
  float s = -1.0f;                                  // removes exp(0) self term
#pragma unroll
  for (int c = 0; c < NCHUNK_; ++c)
    s += part[(size_t)(f * NCHUNK_ + c) * B_ + j];
  out[(size_t)j * OUTSTR_ + INF_ + f] = s;
}

// ---------------------------------------------------------------------------
// Kernel 4: column std (ddof=1) of x, mean over columns, broadcast to col 576.
// ---------------------------------------------------------------------------
__global__ __launch_bounds__(512) void std_kernel(
    const float* __restrict__ x, float* __restrict__ out) {
  const int c = threadIdx.x;
  float sum = 0.f, sumsq = 0.f;
  for (int r = 0; r < B_; ++r) {
    const float v = x[(size_t)r * INF_ + c];   // coalesced across lanes
    sum += v;
    sumsq = fmaf(v, v, sumsq);
  }
  const float var = (sumsq - sum * sum * (1.0f / B_)) * (1.0f / (B_ - 1));
  const float sd  = sqrtf(fmaxf(var, 0.f));

  __shared__ float red[512];
  red[c] = sd;
  __syncthreads();
#pragma unroll
  for (int s = 256; s > 0; s >>= 1) {
    if (c < s) red[c] += red[c + s];
    __syncthreads();
  }
  const float mean_std = red[0] * (1.0f / INF_);
  out[(size_t)c * OUTSTR_ + (INF_ + OUTF_)] = mean_std;  // row c, col 576
}

// ---------------------------------------------------------------------------
// Kernel 5: copy x into the first 512 columns of the 577-wide output.
// ---------------------------------------------------------------------------
__global__ __launch_bounds__(256) void copy_x_kernel(
    const float* __restrict__ x, float* __restrict__ out) {
  const int idx = blockIdx.x * 256 + threadIdx.x;  // 0 .. 512*512-1
  const int i = idx >> 9, c = idx & 511;
  out[(size_t)i * OUTSTR_ + c] = x[idx];
}

extern "C" void kernel_launch(void* const* d_in, const int* in_sizes, int n_in,
                              void* d_out, int out_size, void* d_ws, size_t ws_size,
                              hipStream_t stream) {
  (void)in_sizes; (void)n_in; (void)out_size; (void)ws_size;
  const float* x = (const float*)d_in[0];   // [512, 512] f32
  const float* T = (const float*)d_in[1];   // [512, 64, 16] f32
  float* out = (float*)d_out;               // [512, 577] f32
  float* Mw   = (float*)d_ws;                         // [64][512][16] = 2 MB
  float* part = Mw + (size_t)OUTF_ * B_ * KD_;        // [512][512]    = 1 MB

  gemm_wmma_kernel<<<256, 256, 0, stream>>>(x, T, Mw);
  copy_x_kernel<<<(B_ * INF_) / 256, 256, 0, stream>>>(x, out);
  std_kernel<<<1, 512, 0, stream>>>(x, out);
  sim_part_kernel<<<OUTF_ * NCHUNK_, 512, 0, stream>>>(Mw, part);
  sim_reduce_kernel<<<(OUTF_ * B_) / 256, 256, 0, stream>>>(part, out);
}